// LossEvaluation_36386962932176
// MI455X (gfx1250) — compile-verified
//
#include <hip/hip_runtime.h>
#include <math.h>

// ---------------------------------------------------------------------------
// LossEvaluation for MI455X (gfx1250, wave32).
// Distance matrices (64x64 pairwise 2-D squared distances) are computed with
// V_WMMA_F32_16X16X4_F32 using the augmented-Gram trick:
//   A row i = (-2ax, -2ay, |a|^2, 1),  B col j = (bx, by, 1, |b|^2)
//   => D_ij = |a_i - b_j|^2  in a single WMMA per 16x16 tile.
// Collision term prunes the 65536-cell map to the ~330 occupied cells first.
// ---------------------------------------------------------------------------

typedef __attribute__((ext_vector_type(2))) float v2f;
typedef __attribute__((ext_vector_type(8))) float v8f;

#define RES_F     0.05f
#define MAP_THR_F 128.0f
#define BIG2      1.0e18f   // "BIG" in squared-distance space

__device__ __forceinline__ float warpSum32(float x) {
#pragma unroll
  for (int off = 16; off >= 1; off >>= 1) x += __shfl_xor(x, off, 32);
  return x;
}

// Average-Hausdorff between two 64-point 2D sets; one wave per call.
// Returns the same value on every lane.
__device__ float hausdorff64(const float* __restrict__ A,
                             const float* __restrict__ B,
                             bool useAbs) {
  const int  lane = threadIdx.x & 31;
  const int  idx  = lane & 15;
  const bool hi   = (lane & 16) != 0;

  // Preload the four 4x16 B-matrix tiles (2 VGPRs each).
  v2f bt[4];
#pragma unroll
  for (int nt = 0; nt < 4; ++nt) {
    int   j  = nt * 16 + idx;
    float bx = B[2 * j], by = B[2 * j + 1];
    if (useAbs) { bx = fabsf(bx); by = fabsf(by); }
    float n2 = bx * bx + by * by;
    bt[nt].x = hi ? 1.0f : bx;   // K=2 row of B is all ones
    bt[nt].y = hi ? n2   : by;   // K=3 row of B is |b|^2
  }

  float colmin[4] = {BIG2, BIG2, BIG2, BIG2};
  float rowSum = 0.0f;

#pragma unroll
  for (int mt = 0; mt < 4; ++mt) {
    int   i  = mt * 16 + idx;
    float ax = A[2 * i], ay = A[2 * i + 1];
    if (useAbs) { ax = fabsf(ax); ay = fabsf(ay); }
    float a2 = ax * ax + ay * ay;
    v2f at;
    at.x = hi ? a2   : -2.0f * ax;   // K=2 column of A is |a|^2
    at.y = hi ? 1.0f : -2.0f * ay;   // K=3 column of A is ones

    v8f rowmin;
#pragma unroll
    for (int r = 0; r < 8; ++r) rowmin[r] = BIG2;

#pragma unroll
    for (int nt = 0; nt < 4; ++nt) {
      v8f c = {};
      // D = A(16x4) * B(4x16) + 0  ->  16x16 tile of squared distances
      v8f d2 = __builtin_amdgcn_wmma_f32_16x16x4_f32(
          false, at, false, bt[nt], (short)0, c, false, false);
      float cm = BIG2;
#pragma unroll
      for (int r = 0; r < 8; ++r) {
        float v   = d2[r];
        rowmin[r] = fminf(rowmin[r], v);
        cm        = fminf(cm, v);
      }
      // combine the two row-halves (M=r vs M=r+8) of this column
      cm = fminf(cm, __shfl_xor(cm, 16, 32));
      colmin[nt] = fminf(colmin[nt], cm);
    }

    // Butterfly across the 16-lane half-wave: full row minima over all 64 cols
#pragma unroll
    for (int off = 1; off < 16; off <<= 1) {
#pragma unroll
      for (int r = 0; r < 8; ++r)
        rowmin[r] = fminf(rowmin[r], __shfl_xor(rowmin[r], off, 32));
    }
#pragma unroll
    for (int r = 0; r < 8; ++r)
      rowSum += sqrtf(fmaxf(rowmin[r], 0.0f));   // rows mt*16 + 8*hi + r
  }

  float colSum = 0.0f;
#pragma unroll
  for (int nt = 0; nt < 4; ++nt)
    colSum += sqrtf(fmaxf(colmin[nt], 0.0f));

  // each row's min appears on 16 lanes; each column's min on 2 lanes
  float totRow = warpSum32(rowSum) * (1.0f / 16.0f);
  float totCol = warpSum32(colSum) * 0.5f;
  return 0.5f * (totRow + totCol) * (1.0f / 64.0f);
}

// ---------------------------------------------------------------------------

__global__ void k_zero(int* cnt) {
  if (threadIdx.x < 8) cnt[threadIdx.x] = 0;
}

// Compact occupied map cells: grid = 8*256 blocks of 256 threads.
__global__ void k_compact(const float* __restrict__ lmap, int* cnt,
                          unsigned* __restrict__ occ) {
  int b    = blockIdx.x >> 8;
  int cell = ((blockIdx.x & 255) << 8) | threadIdx.x;   // i = cell>>8, j = cell&255
  if (lmap[(size_t)b * 65536 + cell] > 0.0f) {
    int pos = atomicAdd(&cnt[b], 1);
    occ[(size_t)b * 65536 + pos] = (unsigned)cell;
  }
}

// dists[b][n] = hausdorff(yhat[b,n], ygt[b]);  grid = 128 waves.
__global__ void k_dists(const float* __restrict__ yhat,
                        const float* __restrict__ ygt,
                        float* __restrict__ dists) {
  int task = blockIdx.x;              // b*16 + n
  int b    = task >> 4;
  const float* A = yhat + (size_t)task * 128;
  const float* B = ygt + (size_t)b * 128;
  float h = hausdorff64(A, B, false);
  if (threadIdx.x == 0) dists[task] = h;
}

// pair_d / pair_d_abs over triu(16,1) pairs;  grid = 8*120 waves.
__global__ void k_pairs(const float* __restrict__ yhat,
                        float* __restrict__ pair_d,
                        float* __restrict__ pair_abs) {
  int task = blockIdx.x;              // b*120 + p
  int b    = task / 120;
  int p    = task - b * 120;
  int x = 0, rem = p;
  while (rem >= 15 - x) { rem -= 15 - x; ++x; }   // row-major triu order
  int y = x + 1 + rem;
  const float* A = yhat + (size_t)(b * 16 + x) * 128;
  const float* B = yhat + (size_t)(b * 16 + y) * 128;
  float h  = hausdorff64(A, B, false);
  float ha = hausdorff64(A, B, true);
  if (threadIdx.x == 0) { pair_d[task] = h; pair_abs[task] = ha; }
}

// Collision: grid = 128 blocks (one per (b,n) path), 256 threads.
__global__ void k_coll(const float* __restrict__ yhat,
                       const int* __restrict__ cnt,
                       const unsigned* __restrict__ occ,
                       float* __restrict__ coll) {
  __shared__ float2 path[64];
  __shared__ float  red[256];
  int task = blockIdx.x;
  int b    = task >> 4;
  int t    = threadIdx.x;
  if (t < 64) {
    float px = yhat[((size_t)task * 64 + t) * 2 + 0] / RES_F + MAP_THR_F;
    float py = yhat[((size_t)task * 64 + t) * 2 + 1] / RES_F + MAP_THR_F;
    path[t] = make_float2(px, py);
  }
  __syncthreads();
  int n = cnt[b];
  const unsigned* op = occ + (size_t)b * 65536;
  float best = BIG2;                       // min squared distance
  for (int s = 0; s < 63; ++s) {
    float2 p1 = path[s], p2 = path[s + 1];
    float vx = p2.x - p1.x, vy = p2.y - p1.y;
    float c2 = vx * vx + vy * vy;
    for (int k = t; k < n; k += 256) {
      unsigned cell = op[k];
      float qx = (float)(cell >> 8), qy = (float)(cell & 255u);
      float wx = qx - p1.x, wy = qy - p1.y;
      float tt = (wx * vx + wy * vy) / c2;
      tt = fminf(fmaxf(tt, 0.0f), 1.0f);
      float dx = wx - tt * vx, dy = wy - tt * vy;
      best = fminf(best, dx * dx + dy * dy);
    }
  }
  red[t] = best;
  __syncthreads();
  for (int off = 128; off >= 1; off >>= 1) {
    if (t < off) red[t] = fminf(red[t], red[t + off]);
    __syncthreads();
  }
  if (t == 0) {
    float dmin = sqrtf(red[0]);
    float dp   = 1.0f - fminf(fmaxf(dmin * RES_F, 1.0e-4f), 0.999f);
    coll[task] = atanhf(dp);
  }
}

// Single-block final reduction.
__global__ void k_final(const float* __restrict__ yhat,
                        const float* __restrict__ y_last,
                        const float* __restrict__ dists,
                        const float* __restrict__ pair_d,
                        const float* __restrict__ pair_abs,
                        const float* __restrict__ coll,
                        float* __restrict__ out) {
  __shared__ float red[256];
  __shared__ float red2[256];
  __shared__ float bmin[8];
  __shared__ int   bidx[8];
  int t = threadIdx.x;
  if (t < 8) {
    float mn = dists[t * 16];
    int   mi = 0;
    for (int n = 1; n < 16; ++n) {
      float v = dists[t * 16 + n];
      if (v < mn) { mn = v; mi = n; }          // first-min, like jnp.argmin
    }
    bmin[t] = mn;
    bidx[t] = mi;
  }
  float s1 = 0.0f, s2 = 0.0f;
  for (int k = t; k < 960; k += 256) {
    float pd = pair_d[k];
    s1 += pd;
    s2 += expf(pd - pair_abs[k]) - 1.0f;
  }
  float s3 = 0.0f, m3 = -1.0e30f;
  if (t < 128) { s3 = coll[t]; m3 = coll[t]; }

  red[t] = s1; red2[t] = s2;
  __syncthreads();
  for (int off = 128; off >= 1; off >>= 1) {
    if (t < off) { red[t] += red[t + off]; red2[t] += red2[t + off]; }
    __syncthreads();
  }
  float pairMean = red[0] * (1.0f / 960.0f);
  float asymMean = red2[0] * (1.0f / 960.0f);
  __syncthreads();
  red[t] = s3; red2[t] = m3;
  __syncthreads();
  for (int off = 128; off >= 1; off >>= 1) {
    if (t < off) { red[t] += red[t + off]; red2[t] = fmaxf(red2[t], red2[t + off]); }
    __syncthreads();
  }
  if (t == 0) {
    float collMean = red[0] * (1.0f / 128.0f);
    float collMax  = red2[0];
    float dl = 0.0f, lpl = 0.0f;
    for (int b = 0; b < 8; ++b) {
      dl += bmin[b];
      int   idx = bidx[b];
      float fx  = yhat[(((size_t)(b * 16 + idx)) * 64 + 63) * 2 + 0];
      float fy  = yhat[(((size_t)(b * 16 + idx)) * 64 + 63) * 2 + 1];
      float dx  = fx - y_last[b * 2 + 0];
      float dy  = fy - y_last[b * 2 + 1];
      lpl += dx * dx + dy * dy;
    }
    dl  *= (1.0f / 8.0f);
    lpl *= (1.0f / 16.0f);
    float diversity = expf(-pairMean);
    out[0] = dl + lpl + diversity + 0.5f * asymMean + collMean + 0.5f * collMax;
  }
}

// ---------------------------------------------------------------------------

extern "C" void kernel_launch(void* const* d_in, const int* in_sizes, int n_in,
                              void* d_out, int out_size, void* d_ws, size_t ws_size,
                              hipStream_t stream) {
  const float* yhat      = (const float*)d_in[0];   // (8,16,64,2)
  const float* ygt       = (const float*)d_in[1];   // (8,64,2)
  const float* y_last    = (const float*)d_in[2];   // (8,2)
  const float* local_map = (const float*)d_in[3];   // (8,256,256)
  float* out = (float*)d_out;

  // Workspace layout (~2.01 MB total)
  char*     ws       = (char*)d_ws;
  int*      cnt      = (int*)ws;                         // 8 ints
  float*    dists    = (float*)(ws + 64);                // 128
  float*    pair_d   = (float*)(ws + 64 + 512);          // 960
  float*    pair_abs = (float*)(ws + 64 + 512 + 3840);   // 960
  float*    coll     = (float*)(ws + 64 + 512 + 7680);   // 128
  unsigned* occ      = (unsigned*)(ws + 64 + 512 + 7680 + 512);  // 8*65536 u32

  k_zero<<<1, 32, 0, stream>>>(cnt);
  k_compact<<<8 * 256, 256, 0, stream>>>(local_map, cnt, occ);
  k_dists<<<128, 32, 0, stream>>>(yhat, ygt, dists);
  k_pairs<<<8 * 120, 32, 0, stream>>>(yhat, pair_d, pair_abs);
  k_coll<<<128, 256, 0, stream>>>(yhat, cnt, occ, coll);
  k_final<<<1, 256, 0, stream>>>(yhat, y_last, dists, pair_d, pair_abs, coll, out);
}